// GaussianVoxelizer_87162066305341
// MI455X (gfx1250) — compile-verified
//
#include <hip/hip_runtime.h>

typedef float v2f __attribute__((ext_vector_type(2)));
typedef float v8f __attribute__((ext_vector_type(8)));

#define GRID 64
#define NVOX (GRID * GRID * GRID)

// ---------------------------------------------------------------------------
// Stage 1: per-gaussian coefficient vector W[g][12] for the feature GEMM
//   qf' = -0.5*log2(e)*qf + log2(dens), features:
//   [x^2, y^2, z^2, xy, xz, yz, x, y, z, 1, 0, 0]
// Also writes radii (int32) after the volume in d_out.
// ---------------------------------------------------------------------------
__global__ __launch_bounds__(256) void gv_prep(
    const float* __restrict__ means, const float* __restrict__ dens,
    const float* __restrict__ scales, const float* __restrict__ rots,
    float* __restrict__ W, int* __restrict__ radii, int N)
{
    int g = blockIdx.x * blockDim.x + threadIdx.x;
    if (g >= N) return;

    float qw = rots[g * 4 + 0], qx = rots[g * 4 + 1];
    float qy = rots[g * 4 + 2], qz = rots[g * 4 + 3];
    float inv = __frsqrt_rn(qw * qw + qx * qx + qy * qy + qz * qz);
    qw *= inv; qx *= inv; qy *= inv; qz *= inv;

    float r00 = 1.f - 2.f * (qy * qy + qz * qz), r01 = 2.f * (qx * qy - qw * qz), r02 = 2.f * (qx * qz + qw * qy);
    float r10 = 2.f * (qx * qy + qw * qz), r11 = 1.f - 2.f * (qx * qx + qz * qz), r12 = 2.f * (qy * qz - qw * qx);
    float r20 = 2.f * (qx * qz - qw * qy), r21 = 2.f * (qy * qz + qw * qx), r22 = 1.f - 2.f * (qx * qx + qy * qy);

    float s0 = scales[g * 3 + 0], s1 = scales[g * 3 + 1], s2 = scales[g * 3 + 2];
    float i0 = 1.f / (s0 * s0), i1 = 1.f / (s1 * s1), i2 = 1.f / (s2 * s2);

    // A = R * diag(i) * R^T  (symmetric)
    float a00 = r00 * i0 * r00 + r01 * i1 * r01 + r02 * i2 * r02;
    float a01 = r00 * i0 * r10 + r01 * i1 * r11 + r02 * i2 * r12;
    float a02 = r00 * i0 * r20 + r01 * i1 * r21 + r02 * i2 * r22;
    float a11 = r10 * i0 * r10 + r11 * i1 * r11 + r12 * i2 * r12;
    float a12 = r10 * i0 * r20 + r11 * i1 * r21 + r12 * i2 * r22;
    float a22 = r20 * i0 * r20 + r21 * i1 * r21 + r22 * i2 * r22;

    float mx = means[g * 3 + 0], my = means[g * 3 + 1], mz = means[g * 3 + 2];
    float Amx = a00 * mx + a01 * my + a02 * mz;
    float Amy = a01 * mx + a11 * my + a12 * mz;
    float Amz = a02 * mx + a12 * my + a22 * mz;
    float bx = -2.f * Amx, by = -2.f * Amy, bz = -2.f * Amz;
    float c0 = mx * Amx + my * Amy + mz * Amz;

    const float K = -0.72134752044448170368f;   // -0.5 * log2(e)
    float* w = W + (size_t)g * 12;
    w[0] = K * a00;        w[1] = K * a11;        w[2]  = K * a22;
    w[3] = K * 2.f * a01;  w[4] = K * 2.f * a02;  w[5]  = K * 2.f * a12;
    w[6] = K * bx;         w[7] = K * by;         w[8]  = K * bz;
    w[9] = K * c0 + __builtin_amdgcn_logf(dens[g]);   // v_log_f32 == log2
    w[10] = 0.f;           w[11] = 0.f;

    float smax = fmaxf(s0, fmaxf(s1, s2));
    radii[g] = (int)ceilf(3.0f * smax * (float)GRID);
}

// ---------------------------------------------------------------------------
// Stage 2: feature-GEMM voxelizer. One wave = TWO 16-voxel tiles (two
// loop-invariant B operand sets + two accumulators) against all gaussians in
// tiles of 16 (A, streamed once per iteration and reused for both B tiles).
// 6x V_WMMA_F32_16X16X4_F32 per iteration (K=12 incl. padding), then
// per-lane exp2 + accumulate. Amortizes A-operand vmem 2x per exp.
// 32-bit 16x4 A / 4x16 B layout: lanes 0-15 carry K={0,1}, lanes 16-31 K={2,3}.
// C/D layout: lane l holds N = l%16, M = vgpr + 8*(l/16).
// ---------------------------------------------------------------------------
__device__ __forceinline__ void make_b(int vid, int kh, v2f& b0, v2f& b1, v2f& b2)
{
    const float vs = 1.0f / (float)GRID;
    float x = ((float)(vid & (GRID - 1)) + 0.5f) * vs;
    float y = ((float)((vid >> 6) & (GRID - 1)) + 0.5f) * vs;
    float z = ((float)(vid >> 12) + 0.5f) * vs;
    float f0 = x * x, f1 = y * y, f2 = z * z;
    float f3 = x * y, f4 = x * z, f5 = y * z;
    b0.x = kh ? f2  : f0;   b0.y = kh ? f3  : f1;   // K 0..3
    b1.x = kh ? x   : f4;   b1.y = kh ? y   : f5;   // K 4..7
    b2.x = kh ? 0.f : z;    b2.y = kh ? 0.f : 1.f;  // K 8..11 (10,11 pad)
}

__device__ __forceinline__ float tile_exp_sum(const v8f& c)
{
    float e0 = __builtin_amdgcn_exp2f(c[0]);
    float e1 = __builtin_amdgcn_exp2f(c[1]);
    float e2 = __builtin_amdgcn_exp2f(c[2]);
    float e3 = __builtin_amdgcn_exp2f(c[3]);
    float e4 = __builtin_amdgcn_exp2f(c[4]);
    float e5 = __builtin_amdgcn_exp2f(c[5]);
    float e6 = __builtin_amdgcn_exp2f(c[6]);
    float e7 = __builtin_amdgcn_exp2f(c[7]);
    return ((e0 + e1) + (e2 + e3)) + ((e4 + e5) + (e6 + e7));
}

__global__ __launch_bounds__(256) void gv_wmma(
    const float* __restrict__ W, float* __restrict__ vol, int N)
{
    const int lane = threadIdx.x & 31;
    const int wave = threadIdx.x >> 5;
    const int wgl  = blockIdx.x * (blockDim.x >> 5) + wave;   // global wave id
    const int n    = lane & 15;       // voxel within tile / gaussian row
    const int kh   = lane >> 4;       // K half-select

    const int tile0 = wgl * 2;        // two consecutive 16-voxel tiles per wave
    const int vid0  = tile0 * 16 + n;
    const int vid1  = vid0 + 16;

    v2f b0a, b1a, b2a, b0b, b1b, b2b; // voxel features, loop-invariant
    make_b(vid0, kh, b0a, b1a, b2a);
    make_b(vid1, kh, b0b, b1b, b2b);

    const float* wp = W + (size_t)n * 12 + 2 * kh;
    float acc0 = 0.f, acc1 = 0.f;
    const int niter = N >> 4;

    #pragma unroll 2
    for (int it = 0; it < niter; ++it) {
        v2f a0 = *(const v2f*)(wp + 0);
        v2f a1 = *(const v2f*)(wp + 4);
        v2f a2 = *(const v2f*)(wp + 8);
        wp += 16 * 12;

        v8f c0 = {};
        c0 = __builtin_amdgcn_wmma_f32_16x16x4_f32(false, a0, false, b0a, (short)0, c0, false, false);
        c0 = __builtin_amdgcn_wmma_f32_16x16x4_f32(false, a1, false, b1a, (short)0, c0, false, false);
        c0 = __builtin_amdgcn_wmma_f32_16x16x4_f32(false, a2, false, b2a, (short)0, c0, false, false);

        v8f c1 = {};
        c1 = __builtin_amdgcn_wmma_f32_16x16x4_f32(false, a0, false, b0b, (short)0, c1, false, false);
        c1 = __builtin_amdgcn_wmma_f32_16x16x4_f32(false, a1, false, b1b, (short)0, c1, false, false);
        c1 = __builtin_amdgcn_wmma_f32_16x16x4_f32(false, a2, false, b2b, (short)0, c1, false, false);

        acc0 += tile_exp_sum(c0);
        acc1 += tile_exp_sum(c1);
    }

    // lanes l and l+16 hold the two M-halves of voxel n: combine.
    acc0 += __shfl_xor(acc0, 16, 32);
    acc1 += __shfl_xor(acc1, 16, 32);
    if (kh == 0) {
        vol[vid0] = acc0;
        vol[vid1] = acc1;
    }
}

// ---------------------------------------------------------------------------
extern "C" void kernel_launch(void* const* d_in, const int* in_sizes, int n_in,
                              void* d_out, int out_size, void* d_ws, size_t ws_size,
                              hipStream_t stream) {
    const float* means  = (const float*)d_in[0];
    const float* dens   = (const float*)d_in[1];
    const float* scales = (const float*)d_in[2];
    const float* rots   = (const float*)d_in[3];
    const int N = in_sizes[1];                    // 4096 gaussians

    float* vol   = (float*)d_out;                 // GRID^3 floats
    int*   radii = (int*)d_out + NVOX;            // N int32 after volume
    float* W     = (float*)d_ws;                  // N*12 floats (192 KB)

    gv_prep<<<(N + 255) / 256, 256, 0, stream>>>(means, dens, scales, rots, W, radii, N);

    const int tiles  = NVOX / 16;                 // 16384 voxel tiles
    const int waves  = tiles / 2;                 // 2 tiles per wave
    const int blocks = waves / 8;                 // 8 waves (256 thr) per block
    gv_wmma<<<blocks, 256, 0, stream>>>(W, vol, N);
}